// RnCLoss_4355096838960
// MI455X (gfx1250) — compile-verified
//
#include <hip/hip_runtime.h>
#include <hip/hip_bf16.h>

typedef __attribute__((ext_vector_type(2))) float v2f;
typedef __attribute__((ext_vector_type(8))) float v8f;

constexpr int NN = 384;   // rows
constexpr int DD = 256;   // feature dim
constexpr float INV_TEMP = 0.5f;   // 1 / TEMPERATURE(=2)

// ---------------------------------------------------------------------------
// Kernel 1: row squared-norms  ||f_i||^2
// ---------------------------------------------------------------------------
__global__ __launch_bounds__(256) void row_norms_kernel(const float* __restrict__ F,
                                                        float* __restrict__ norms) {
    __shared__ float red[256];
    const int i = blockIdx.x;
    const int t = threadIdx.x;
    const float v = F[i * DD + t];
    red[t] = v * v;
    __syncthreads();
    for (int s = 128; s > 0; s >>= 1) {
        if (t < s) red[t] += red[t + s];
        __syncthreads();
    }
    if (t == 0) norms[i] = red[0];
}

// ---------------------------------------------------------------------------
// Kernel 2: Gram tile via f32 WMMA -> logits = -dist/T and exp(logits)
// One wave32 per 16x16 output tile; K-loop in steps of 4 (V_WMMA_F32_16X16X4_F32).
// A-matrix layout (ISA 7.12.2, 32-bit A 16x4): lane&15 = row M, lane>=16 holds K=2,3.
// B = F^T, so the B-tile load is the same expression with the column index.
// ---------------------------------------------------------------------------
__global__ __launch_bounds__(32) void gram_logits_kernel(const float* __restrict__ F,
                                                         const float* __restrict__ norms,
                                                         float* __restrict__ lg,
                                                         float* __restrict__ ex) {
    const int bi   = blockIdx.x;       // row tile
    const int bj   = blockIdx.y;       // col tile
    const int lane = threadIdx.x;      // 0..31
    const int half = lane >> 4;        // 0: K=0,1   1: K=2,3
    const int l15  = lane & 15;

    const float* pa = F + (bi * 16 + l15) * DD + half * 2;  // A: row of F
    const float* pb = F + (bj * 16 + l15) * DD + half * 2;  // B: col of F^T = row of F

    v8f acc = {};
    #pragma unroll 4
    for (int k = 0; k < DD; k += 4) {
        v2f a = *(const v2f*)(pa + k);
        v2f b = *(const v2f*)(pb + k);
        // 8 args: (neg_a, A, neg_b, B, c_mod, C, reuse_a, reuse_b)
        acc = __builtin_amdgcn_wmma_f32_16x16x4_f32(false, a, false, b,
                                                    (short)0, acc, false, false);
    }

    // C/D layout: VGPR r, lanes 0-15 -> M=r, lanes 16-31 -> M=8+r; N = lane&15.
    const int   n  = bj * 16 + l15;
    const float nj = norms[n];
    #pragma unroll
    for (int r = 0; r < 8; ++r) {
        const int   m  = bi * 16 + (half ? (8 + r) : r);
        float d2 = norms[m] + nj - 2.0f * acc[r];
        d2 = fmaxf(d2, 0.0f);
        const float dist  = sqrtf(d2);           // d2==0 -> dist 0, matches safe-sqrt path
        const float logit = -dist * INV_TEMP;    // row-max subtraction is exactly 0
        lg[m * NN + n] = logit;
        ex[m * NN + n] = expf(logit);
    }
}

// ---------------------------------------------------------------------------
// Kernel 3: per-row  sum_j [ logits(i,j) - log(denom(i,j)) ],  j != i
// denom(i,j) = sum_{k != i, ld(i,k) >= ld(i,j)} exp_logits(i,k)
// ---------------------------------------------------------------------------
__global__ __launch_bounds__(128) void denom_kernel(const float* __restrict__ labels,
                                                    const float* __restrict__ lg,
                                                    const float* __restrict__ ex,
                                                    float* __restrict__ rowsum) {
    __shared__ float sE[NN];
    __shared__ float sLD[NN];
    __shared__ float red[128];

    const int i  = blockIdx.x;
    const int t  = threadIdx.x;
    const float yi = labels[i];

    for (int k = t; k < NN; k += 128) {
        sE[k]  = ex[i * NN + k];
        sLD[k] = fabsf(yi - labels[k]);
    }
    __syncthreads();

    float local = 0.0f;
    for (int j = t; j < NN; j += 128) {
        if (j == i) continue;
        const float ldj = sLD[j];
        float denom = 0.0f;
        for (int k = 0; k < NN; ++k) {
            const float e = (k != i && sLD[k] >= ldj) ? sE[k] : 0.0f;
            denom += e;
        }
        local += lg[i * NN + j] - logf(denom);
    }

    red[t] = local;
    __syncthreads();
    for (int s = 64; s > 0; s >>= 1) {
        if (t < s) red[t] += red[t + s];
        __syncthreads();
    }
    if (t == 0) rowsum[i] = red[0];
}

// ---------------------------------------------------------------------------
// Kernel 4: final reduction -> scalar loss
// ---------------------------------------------------------------------------
__global__ __launch_bounds__(384) void finalize_kernel(const float* __restrict__ rowsum,
                                                       float* __restrict__ out) {
    __shared__ float red[384];
    const int t = threadIdx.x;
    red[t] = rowsum[t];
    __syncthreads();
    if (t < 128) red[t] += red[t + 256];   // fold 384 -> 256
    __syncthreads();
    for (int s = 128; s > 0; s >>= 1) {
        if (t < s) red[t] += red[t + s];
        __syncthreads();
    }
    if (t == 0) out[0] = -red[0] / (float)(NN * (NN - 1));
}

// ---------------------------------------------------------------------------
extern "C" void kernel_launch(void* const* d_in, const int* in_sizes, int n_in,
                              void* d_out, int out_size, void* d_ws, size_t ws_size,
                              hipStream_t stream) {
    (void)in_sizes; (void)n_in; (void)out_size; (void)ws_size;

    const float* F      = (const float*)d_in[0];   // [384, 256] f32
    const float* labels = (const float*)d_in[1];   // [384, 1]   f32
    float* out = (float*)d_out;                    // scalar f32

    float* ws     = (float*)d_ws;
    float* norms  = ws;                    // 384   (padded to 512)
    float* lg     = ws + 512;              // 384*384 logits
    float* ex     = lg + NN * NN;          // 384*384 exp(logits)
    float* rowsum = ex + NN * NN;          // 384 partials

    row_norms_kernel<<<NN, 256, 0, stream>>>(F, norms);

    dim3 grid(NN / 16, NN / 16);
    gram_logits_kernel<<<grid, 32, 0, stream>>>(F, norms, lg, ex);

    denom_kernel<<<NN, 128, 0, stream>>>(labels, lg, ex, rowsum);

    finalize_kernel<<<1, 384, 0, stream>>>(rowsum, out);
}